// TokenEmbedding_29549374996902
// MI455X (gfx1250) — compile-verified
//
#include <hip/hip_runtime.h>
#include <hip/hip_bf16.h>

typedef __attribute__((ext_vector_type(2))) float v2f;
typedef __attribute__((ext_vector_type(8))) float v8f;

#define TAO     24
#define MLAG    7
#define C_IN    7
#define D_MODEL 512
#define SEQ     2048
#define KDIM    168   // 56 * 3
#define NSTEPS  42    // KDIM / 4
#define ASTRIDE 172   // padded LDS row stride (floats)

// out[b,t,o] = bias[o] + sum_{i,k} W[o,i,k] * z[b,i,(t+k-1) mod 2048]
// z[b, c*8+m, s] = (s >= 168) ? x[b, s-24m, c] : 0
__global__ __launch_bounds__(256)
void tdc_conv_wmma_kernel(const float* __restrict__ x,
                          const float* __restrict__ W,
                          const float* __restrict__ bias,
                          float* __restrict__ out)
{
    __shared__ float Atile[16 * ASTRIDE];

    const int block = blockIdx.x;        // B * (SEQ/16) = 32*128 = 4096 blocks
    const int batch = block >> 7;        // 128 m-tiles per batch
    const int t0    = (block & 127) << 4;
    const int tid   = threadIdx.x;

    // ---- Stage A tile (16 rows x 168 K) into LDS ----
    // j = i*3 + k ; i = c*8 + m
    for (int e = tid; e < 16 * KDIM; e += 256) {
        const int m_row = e / KDIM;
        const int j     = e - m_row * KDIM;
        const int k     = j % 3;
        const int i     = j / 3;
        const int c     = i >> 3;
        const int mm    = i & 7;
        const int t     = t0 + m_row;
        const int s     = (t + k - 1 + SEQ) & (SEQ - 1);   // wrap pad
        float v = 0.0f;
        if (s >= MLAG * TAO) {
            v = x[((size_t)batch * SEQ + (s - TAO * mm)) * C_IN + c];
        }
        Atile[m_row * ASTRIDE + j] = v;
    }
    __syncthreads();

    const int wave = tid >> 5;    // 0..7 -> owns N-tiles wave*4 .. wave*4+3
    const int lane = tid & 31;
    const int half = lane >> 4;   // selects K pair (A) / K pair (B) / M half (D)
    const int lid  = lane & 15;

    v8f acc0 = {}, acc1 = {}, acc2 = {}, acc3 = {};

    // A operand per lane: A[lid][k0 + 2*half + {0,1}]
    const float* Arow = &Atile[lid * ASTRIDE + 2 * half];
    // B operand per lane: Wf[k0 + 2*half + {0,1}, o] = W[o*KDIM + k0 + 2*half + {0,1}]
    const int o_base = wave * 64;
    const float* Wl0 = W + (size_t)(o_base +  0 + lid) * KDIM + 2 * half;
    const float* Wl1 = W + (size_t)(o_base + 16 + lid) * KDIM + 2 * half;
    const float* Wl2 = W + (size_t)(o_base + 32 + lid) * KDIM + 2 * half;
    const float* Wl3 = W + (size_t)(o_base + 48 + lid) * KDIM + 2 * half;

    #pragma unroll 6
    for (int kk = 0; kk < NSTEPS; ++kk) {
        const int k0 = kk * 4;
        v2f a  = *(const v2f*)(Arow + k0);       // ds_load_b64
        v2f b0 = *(const v2f*)(Wl0 + k0);        // global_load_b64 (L2-resident W)
        v2f b1 = *(const v2f*)(Wl1 + k0);
        v2f b2 = *(const v2f*)(Wl2 + k0);
        v2f b3 = *(const v2f*)(Wl3 + k0);
        acc0 = __builtin_amdgcn_wmma_f32_16x16x4_f32(false, a, false, b0, (short)0, acc0, false, false);
        acc1 = __builtin_amdgcn_wmma_f32_16x16x4_f32(false, a, false, b1, (short)0, acc1, false, false);
        acc2 = __builtin_amdgcn_wmma_f32_16x16x4_f32(false, a, false, b2, (short)0, acc2, false, false);
        acc3 = __builtin_amdgcn_wmma_f32_16x16x4_f32(false, a, false, b3, (short)0, acc3, false, false);
    }

    // ---- Epilogue: D VGPR r holds element (M = r + 8*half, N = lid) ----
    const size_t row_base = ((size_t)batch * SEQ + t0 + 8 * half) * D_MODEL;

    {
        const int o = o_base + 0 + lid;
        const float bo = bias[o];
        #pragma unroll
        for (int r = 0; r < 8; ++r)
            out[row_base + (size_t)r * D_MODEL + o] = acc0[r] + bo;
    }
    {
        const int o = o_base + 16 + lid;
        const float bo = bias[o];
        #pragma unroll
        for (int r = 0; r < 8; ++r)
            out[row_base + (size_t)r * D_MODEL + o] = acc1[r] + bo;
    }
    {
        const int o = o_base + 32 + lid;
        const float bo = bias[o];
        #pragma unroll
        for (int r = 0; r < 8; ++r)
            out[row_base + (size_t)r * D_MODEL + o] = acc2[r] + bo;
    }
    {
        const int o = o_base + 48 + lid;
        const float bo = bias[o];
        #pragma unroll
        for (int r = 0; r < 8; ++r)
            out[row_base + (size_t)r * D_MODEL + o] = acc3[r] + bo;
    }
}

extern "C" void kernel_launch(void* const* d_in, const int* in_sizes, int n_in,
                              void* d_out, int out_size, void* d_ws, size_t ws_size,
                              hipStream_t stream) {
    (void)in_sizes; (void)n_in; (void)d_ws; (void)ws_size; (void)out_size;
    const float* x    = (const float*)d_in[0];   // (32, 2048, 7)
    const float* W    = (const float*)d_in[1];   // (512, 56, 3)
    const float* bias = (const float*)d_in[2];   // (512,)
    float* out        = (float*)d_out;           // (32, 2048, 512)

    const int num_blocks = 32 * (SEQ / 16);      // 4096
    tdc_conv_wmma_kernel<<<num_blocks, 256, 0, stream>>>(x, W, bias, out);
}